// gumbel_wrapper_64905545777930
// MI455X (gfx1250) — compile-verified
//
#include <hip/hip_runtime.h>
#include <math.h>

#define NS 8192
#define HD 64
#define G3 192
#define LL 64
#define NEG_INF (-__builtin_inff())

typedef __attribute__((ext_vector_type(2))) float v2f;
typedef __attribute__((ext_vector_type(8))) float v8f;

__device__ __forceinline__ float logaddexpf_(float a, float b) {
  float m = fmaxf(a, b);
  float mn = fminf(a, b);
  if (!__builtin_isfinite(m)) return m;      // (-inf,-inf)->-inf ; +inf->+inf
  return m + log1pf(expf(mn - m));
}

// ---------------- init: h=0, samples=-2, logits0/gumbel0 = -inf except [0]=0
__global__ void k_init(float* __restrict__ h, int* __restrict__ samp,
                       float* __restrict__ lg0, float* __restrict__ gb0) {
  int id = blockIdx.x * blockDim.x + threadIdx.x;   // NS*HD threads
  if (id < NS * HD) { h[id] = 0.f; samp[id] = -2; }
  if (id < NS) {
    lg0[id] = (id == 0) ? 0.f : NEG_INF;
    gb0[id] = (id == 0) ? 0.f : NEG_INF;
  }
}

// ---------------- gi[tok][j] = embed[tok]@Wx + bx  (2 x 192, precomputed once)
__global__ void k_gi(const float* __restrict__ embed, const float* __restrict__ Wx,
                     const float* __restrict__ bx, float* __restrict__ gi) {
  int id = blockIdx.x * blockDim.x + threadIdx.x;   // 2*192 threads
  if (id < 2 * G3) {
    int t = id / G3, j = id % G3;
    float s = bx[j];
    for (int k = 0; k < HD; ++k) s += embed[t * HD + k] * Wx[k * G3 + j];
    gi[id] = s;
  }
}

// ---------------- GRU cell + candidate gumbels. 128 blocks x 128 thr (4 waves).
// Each wave computes a 16x192 gh tile via chained V_WMMA_F32_16X16X4_F32.
__global__ __launch_bounds__(128) void k_gru(
    const float* __restrict__ h, const int* __restrict__ samp,
    const float* __restrict__ gi, const float* __restrict__ Wh,
    const float* __restrict__ bh, const float* __restrict__ Wo,
    const float* __restrict__ bo, const float* __restrict__ lg0,
    const float* __restrict__ gb0, const float* __restrict__ noise_t,
    int pos, float* __restrict__ h2, float* __restrict__ lnout,
    float* __restrict__ key, int* __restrict__ sidx) {
  __shared__ float ghS[64][G3];   // 48 KB
  __shared__ float h2S[64][HD];   // 16 KB
  const int tid = threadIdx.x;
  const int wave = tid >> 5, lane = tid & 31;
  const int m = lane & 15, kh = lane >> 4;
  const int rowBase = blockIdx.x * 64;
  const int arow = rowBase + wave * 16 + m;

  // pull the shared Wh panel (48KB, L2-resident) toward the WGP
  __builtin_prefetch(Wh + (size_t)tid * 96, 0, 3);

  // A fragments (16x4 f32 layout: VGPR0 <- K = kbase, VGPR1 <- K = kbase+1;
  // lanes 16-31 shifted by 2 in K). Built once, reused across 12 N-tiles.
  v2f afr[16];
#pragma unroll
  for (int kk = 0; kk < 16; ++kk) {
    int kbase = kk * 4 + 2 * kh;
    afr[kk].x = h[arow * HD + kbase];
    afr[kk].y = h[arow * HD + kbase + 1];
  }
  // two N-tiles per iteration -> two independent WMMA accumulation chains
#pragma unroll 2
  for (int nt = 0; nt < 12; ++nt) {
    int ncol = nt * 16 + m;
    v8f c = {0.f, 0.f, 0.f, 0.f, 0.f, 0.f, 0.f, 0.f};
#pragma unroll
    for (int kk = 0; kk < 16; ++kk) {
      int kbase = kk * 4 + 2 * kh;
      v2f b;
      b.x = Wh[kbase * G3 + ncol];
      b.y = Wh[(kbase + 1) * G3 + ncol];
      c = __builtin_amdgcn_wmma_f32_16x16x4_f32(false, afr[kk], false, b,
                                                (short)0, c, false, false);
    }
    // C/D layout: VGPR v -> M = v + 8*kh, N = lane&15
#pragma unroll
    for (int v = 0; v < 8; ++v)
      ghS[wave * 16 + v + 8 * kh][ncol] = c[v];
  }
  __syncthreads();

  // gates: 64 rows x 64 units across 128 threads
  for (int it = 0; it < 32; ++it) {
    int p = it * 128 + tid;
    int r = p >> 6, u = p & 63;
    int grow = rowBase + r;
    int tok = (pos == 0) ? 0 : samp[grow * LL + pos - 1];
    tok = tok < 0 ? 0 : (tok > 1 ? 1 : tok);
    float giz = gi[tok * G3 + u];
    float gir = gi[tok * G3 + 64 + u];
    float gin = gi[tok * G3 + 128 + u];
    float ghz = ghS[r][u] + bh[u];
    float ghr = ghS[r][64 + u] + bh[64 + u];
    float ghn = ghS[r][128 + u] + bh[128 + u];
    float z = 1.f / (1.f + expf(-(giz + ghz)));
    float rr = 1.f / (1.f + expf(-(gir + ghr)));
    float nn = tanhf(gin + rr * ghn);
    float hp = h[grow * HD + u];
    float hv = (1.f - z) * nn + z * hp;
    h2S[r][u] = hv;
    h2[grow * HD + u] = hv;
  }
  __syncthreads();

  // logits, log_softmax(2), truncated gumbels: one thread per row
  if (tid < 64) {
    int r = tid, grow = rowBase + r;
    float o0 = bo[0], o1 = bo[1];
    for (int k2 = 0; k2 < HD; ++k2) {
      float hv = h2S[r][k2];
      o0 += hv * Wo[k2 * 2 + 0];
      o1 += hv * Wo[k2 * 2 + 1];
    }
    float mx = fmaxf(o0, o1);
    float lse = mx + logf(expf(o0 - mx) + expf(o1 - mx));
    float lp0 = o0 - lse, lp1 = o1 - lse;
    float l0 = lg0[grow];
    bool alive = __builtin_isfinite(l0);
    float T = alive ? gb0[grow] : 0.f;
    float lb = alive ? l0 : 0.f;
    float ln0 = lb + lp0, ln1 = lb + lp1;
    float g0 = ln0 + noise_t[grow * 2 + 0];
    float g1 = ln1 + noise_t[grow * 2 + 1];
    float Z = fmaxf(g0, g1);
    float gv[2] = {g0, g1};
    float lnv[2] = {ln0, ln1};
#pragma unroll
    for (int d = 0; d < 2; ++d) {
      float gd = gv[d];
      float gt;
      if ((gd - Z) >= 0.f) {
        gt = T;
      } else {
        float lm = log1pf(-expf(gd - Z));     // log(1 - exp(gd - Z))
        gt = -logaddexpf_(-T, lm - gd);
      }
      key[grow * 2 + d] = alive ? gt : NEG_INF;
      lnout[grow * 2 + d] = alive ? lnv[d] : NEG_INF;
      sidx[grow * 2 + d] = grow * 2 + d;
    }
  }
}

// ---------------- hybrid bitonic sort (descending overall), M = 16384
// Local kernel: 8 blocks x 1024 thr, each owns a contiguous 2048-chunk in LDS.
// mergeSize==0  : full local bitonic sort, stages size=2..2048
// mergeSize==S  : local tail of stage S: strides 1024..1 (direction from S)
__global__ __launch_bounds__(1024) void k_sort_local(float* __restrict__ key,
                                                     int* __restrict__ idx,
                                                     int mergeSize) {
  __shared__ float ks[2048];
  __shared__ int vs[2048];
  const int base = blockIdx.x * 2048;
  const int t = threadIdx.x;
  for (int i = t; i < 2048; i += 1024) { ks[i] = key[base + i]; vs[i] = idx[base + i]; }
  __syncthreads();
  if (mergeSize == 0) {
    for (int size = 2; size <= 2048; size <<= 1) {
      for (int stride = size >> 1; stride > 0; stride >>= 1) {
        int lo = ((t / stride) * (stride << 1)) + (t % stride);
        int hi = lo + stride;
        bool desc = (((base + lo) & size) == 0);
        float a = ks[lo], b = ks[hi];
        bool sw = desc ? (a < b) : (a > b);
        if (sw) {
          ks[lo] = b; ks[hi] = a;
          int tv = vs[lo]; vs[lo] = vs[hi]; vs[hi] = tv;
        }
        __syncthreads();
      }
    }
  } else {
    const bool desc = ((base & mergeSize) == 0);   // constant per chunk
    for (int stride = 1024; stride > 0; stride >>= 1) {
      int lo = ((t / stride) * (stride << 1)) + (t % stride);
      int hi = lo + stride;
      float a = ks[lo], b = ks[hi];
      bool sw = desc ? (a < b) : (a > b);
      if (sw) {
        ks[lo] = b; ks[hi] = a;
        int tv = vs[lo]; vs[lo] = vs[hi]; vs[hi] = tv;
      }
      __syncthreads();
    }
  }
  for (int i = t; i < 2048; i += 1024) { key[base + i] = ks[i]; idx[base + i] = vs[i]; }
}

// Global single pass: independent compare-exchanges, no intra-kernel sync.
__global__ void k_sort_gpass(float* __restrict__ key, int* __restrict__ idx,
                             int size, int stride) {
  int p = blockIdx.x * blockDim.x + threadIdx.x;   // M/2 = 8192 pairs
  int lo = ((p / stride) * (stride << 1)) + (p % stride);
  int hi = lo + stride;
  bool desc = ((lo & size) == 0);
  float a = key[lo], b = key[hi];
  bool sw = desc ? (a < b) : (a > b);
  if (sw) {
    key[lo] = b; key[hi] = a;
    int tv = idx[lo]; idx[lo] = idx[hi]; idx[hi] = tv;
  }
}

// ---------------- gather top-N into next-state buffers
__global__ void k_gather(const int* __restrict__ sidx, const float* __restrict__ skey,
                         const int* __restrict__ samp_in, int* __restrict__ samp_out,
                         const float* __restrict__ h2, float* __restrict__ h,
                         const float* __restrict__ lnin, float* __restrict__ lg0,
                         float* __restrict__ gb0, int pos, int last,
                         float* __restrict__ kap) {
  int gid = blockIdx.x * blockDim.x + threadIdx.x;   // NS*64 threads
  int r = gid >> 6, k = gid & 63;
  int src = sidx[r];
  int n = src >> 1, d = src & 1;
  samp_out[r * LL + k] = (k == pos) ? d : samp_in[n * LL + k];
  h[r * HD + k] = h2[n * HD + k];
  if (k == 0) {
    lg0[r] = lnin[src];
    gb0[r] = skey[r];
  }
  if (gid == 0 && last) kap[0] = skey[NS];   // rank-N gumbel = kappa
}

// ---------------- final outputs: samples(float), lp*0.5, w(unnormalized), kappa
__global__ void k_final(const int* __restrict__ samp, const float* __restrict__ lg0,
                        const float* __restrict__ kap, float* __restrict__ out) {
  int gid = blockIdx.x * blockDim.x + threadIdx.x;   // NS*64 threads
  int n = gid >> 6, k = gid & 63;
  out[gid] = (float)samp[n * LL + k];
  if (k == 0) {
    float lp = lg0[n];
    out[NS * LL + n] = lp * 0.5f;
    float kp = kap[0];
    float e = expf(lp);
    float den = -expm1f(-expf(lp - kp));
    float wv = e / den;
    if (__builtin_isnan(wv)) wv = 0.f;
    else if (__builtin_isinf(wv)) wv = (wv > 0.f) ? 3.4028234663852886e38f
                                                  : -3.4028234663852886e38f;
    out[NS * LL + NS + n] = wv;
  }
  if (gid == 0) out[NS * LL + 2 * NS] = kap[0];
}

// deterministic sum (fixed per-thread strided order + LDS tree)
__global__ __launch_bounds__(1024) void k_wsum(const float* __restrict__ wv,
                                               float* __restrict__ wsum) {
  __shared__ float red[1024];
  float s = 0.f;
  for (int i = (int)threadIdx.x; i < NS; i += 1024) s += wv[i];
  red[threadIdx.x] = s;
  __syncthreads();
  for (int off = 512; off > 0; off >>= 1) {
    if ((int)threadIdx.x < off) red[threadIdx.x] += red[threadIdx.x + off];
    __syncthreads();
  }
  if (threadIdx.x == 0) wsum[0] = red[0];
}

__global__ void k_wnorm(float* __restrict__ wv, const float* __restrict__ wsum) {
  int gid = blockIdx.x * blockDim.x + threadIdx.x;
  if (gid < NS) wv[gid] /= wsum[0];
}

extern "C" void kernel_launch(void* const* d_in, const int* in_sizes, int n_in,
                              void* d_out, int out_size, void* d_ws, size_t ws_size,
                              hipStream_t stream) {
  (void)in_sizes; (void)n_in; (void)out_size; (void)ws_size;
  const float* embed = (const float*)d_in[0];
  const float* Wx    = (const float*)d_in[1];
  const float* Wh    = (const float*)d_in[2];
  const float* bx    = (const float*)d_in[3];
  const float* bh    = (const float*)d_in[4];
  const float* Wo    = (const float*)d_in[5];
  const float* bo    = (const float*)d_in[6];
  const float* noise = (const float*)d_in[7];
  float* out = (float*)d_out;

  char* w = (char*)d_ws;
  size_t off = 0;
  auto alloc = [&](size_t bytes) -> void* {
    void* p = w + off;
    off += (bytes + 255) & ~(size_t)255;
    return p;
  };
  float* hA   = (float*)alloc((size_t)NS * HD * 4);
  float* h2   = (float*)alloc((size_t)NS * HD * 4);
  int*   sA   = (int*)alloc((size_t)NS * LL * 4);
  int*   sB   = (int*)alloc((size_t)NS * LL * 4);
  float* lg0  = (float*)alloc((size_t)NS * 4);
  float* gb0  = (float*)alloc((size_t)NS * 4);
  float* lnb  = (float*)alloc((size_t)2 * NS * 4);
  float* key  = (float*)alloc((size_t)2 * NS * 4);
  int*   sidx = (int*)alloc((size_t)2 * NS * 4);
  float* gi   = (float*)alloc((size_t)2 * G3 * 4);
  float* kap  = (float*)alloc(4);
  float* wsum = (float*)alloc(4);

  k_init<<<NS * HD / 256, 256, 0, stream>>>(hA, sA, lg0, gb0);
  k_gi<<<2, 192, 0, stream>>>(embed, Wx, bx, gi);

  int* scur = sA;
  int* snxt = sB;
  for (int t = 0; t < LL; ++t) {
    k_gru<<<NS / 64, 128, 0, stream>>>(hA, scur, gi, Wh, bh, Wo, bo, lg0, gb0,
                                       noise + (size_t)t * NS * 2, t, h2, lnb,
                                       key, sidx);
    // hybrid bitonic sort of 16384 (key, idx), descending
    k_sort_local<<<8, 1024, 0, stream>>>(key, sidx, 0);           // sizes 2..2048
    for (int size = 4096; size <= 16384; size <<= 1) {
      for (int stride = size >> 1; stride >= 2048; stride >>= 1)
        k_sort_gpass<<<32, 256, 0, stream>>>(key, sidx, size, stride);
      k_sort_local<<<8, 1024, 0, stream>>>(key, sidx, size);      // strides <=1024
    }
    k_gather<<<NS * HD / 256, 256, 0, stream>>>(sidx, key, scur, snxt, h2, hA,
                                                lnb, lg0, gb0, t,
                                                (t == LL - 1) ? 1 : 0, kap);
    int* tmp = scur; scur = snxt; snxt = tmp;
  }

  k_final<<<NS * HD / 256, 256, 0, stream>>>(scur, lg0, kap, out);
  k_wsum<<<1, 1024, 0, stream>>>(out + (size_t)NS * LL + NS, wsum);
  k_wnorm<<<NS / 256, 256, 0, stream>>>(out + (size_t)NS * LL + NS, wsum);
}